// hierarchically_modular_37907381355022
// MI455X (gfx1250) — compile-verified
//
#include <hip/hip_runtime.h>
#include <math.h>

// ---------------------------------------------------------------------------
// Problem constants (match reference)
// ---------------------------------------------------------------------------
#define BATCH      131072
#define D_IN       64
#define M0         64
#define M1         64
#define HID        12

// rows per wave = 16 (one WMMA M-tile); 8 waves/block -> 128 rows/block
#define WAVES_PER_BLOCK 8
#define ROWS_PER_BLOCK  (WAVES_PER_BLOCK * 16)
#define NBLOCKS         (BATCH / ROWS_PER_BLOCK)   // 1024

typedef __attribute__((ext_vector_type(2))) float v2f;
typedef __attribute__((ext_vector_type(8))) float v8f;

// Packed routing plan living in d_ws. Only the modules that can influence the
// output survive: 2 layer-1 modules (j0=top1,j1=top2 of emb_out), their 2x2
// layer-0 parents p[0..3], and the 8 x-columns c[0..7] those parents gather.
struct Plan {
  int   c[8];            // x feature columns; c[2u],c[2u+1] = top1/top2 of emb0 col p[u]
  float Wblk[8 * 48];    // block-diagonal layer-0 W1: rows 2u,2u+1 x cols 12u..12u+11
  float b1blk[48];       // layer-0 b1 per column
  float w2blk[48];       // layer-0 W2 per column
  float b2l0[4];         // layer-0 b2 per selected module
  float W1l1[2 * 2 * HID]; // [t][i][h] for layer-1 modules j0,j1
  float b1l1[2 * HID];
  float w2l1[2 * HID];
  float b2l1[2];
};

// ---------------------------------------------------------------------------
// Setup: replay the static routing (jax top_k: ties -> lowest index, so use
// strict '>' scans) and pack the reduced weight set.
// ---------------------------------------------------------------------------
__device__ inline void top2_scan(const float* base, int stride, int& i1, int& i2) {
  float m1 = -3.402823466e38f; int a1 = 0;
  for (int d = 0; d < 64; ++d) {
    float v = base[d * stride];
    if (v > m1) { m1 = v; a1 = d; }
  }
  float m2 = -3.402823466e38f; int a2 = 0;
  for (int d = 0; d < 64; ++d) {
    if (d == a1) continue;
    float v = base[d * stride];
    if (v > m2) { m2 = v; a2 = d; }
  }
  i1 = a1; i2 = a2;
}

__global__ void setup_plan_kernel(const float* __restrict__ emb0,
                                  const float* __restrict__ emb1,
                                  const float* __restrict__ emb_out,
                                  const int*   __restrict__ task_id_p,
                                  const float* __restrict__ W1_0, const float* __restrict__ b1_0,
                                  const float* __restrict__ W2_0, const float* __restrict__ b2_0,
                                  const float* __restrict__ W1_1, const float* __restrict__ b1_1,
                                  const float* __restrict__ W2_1, const float* __restrict__ b2_1,
                                  Plan* __restrict__ plan) {
  __shared__ int jsh[2];   // layer-1 module ids (output cols 0,1)
  __shared__ int psh[4];   // layer-0 module ids
  __shared__ int csh[8];   // x columns
  const int t = threadIdx.x;
  const int task = *task_id_p;
  const float* e0 = emb0 + (size_t)task * D_IN * M0;   // [d*M0 + m]
  const float* e1 = emb1 + (size_t)task * M0 * M1;     // [d*M1 + j]
  const float* eo = emb_out + (size_t)task * M1;       // [d]

  if (t == 0) { top2_scan(eo, 1, jsh[0], jsh[1]); }
  __syncthreads();
  if (t < 2)  { top2_scan(e1 + jsh[t], M1, psh[2 * t], psh[2 * t + 1]); }
  __syncthreads();
  if (t < 4)  { top2_scan(e0 + psh[t], M0, csh[2 * t], csh[2 * t + 1]); }
  __syncthreads();

  if (t < 8) plan->c[t] = csh[t];
  for (int i = t; i < 8 * 48; i += 64) plan->Wblk[i] = 0.0f;
  __syncthreads();

  if (t < 48) {                       // t = u*12 + h
    const int u = t / HID, h = t % HID;
    const int m = psh[u];
    // W1_0: [m][i][h] with i in {0,1}
    plan->Wblk[(2 * u + 0) * 48 + t] = W1_0[m * (2 * HID) + 0 * HID + h];
    plan->Wblk[(2 * u + 1) * 48 + t] = W1_0[m * (2 * HID) + 1 * HID + h];
    plan->b1blk[t] = b1_0[m * HID + h];
    plan->w2blk[t] = W2_0[m * HID + h];
  }
  if (t < 4)  plan->b2l0[t] = b2_0[psh[t]];
  if (t < 2 * HID) {                  // t = tt*12 + h
    const int tt = t / HID, h = t % HID;
    const int m = jsh[tt];
    plan->W1l1[tt * (2 * HID) + 0 * HID + h] = W1_1[m * (2 * HID) + 0 * HID + h];
    plan->W1l1[tt * (2 * HID) + 1 * HID + h] = W1_1[m * (2 * HID) + 1 * HID + h];
    plan->b1l1[t] = b1_1[m * HID + h];
    plan->w2l1[t] = W2_1[m * HID + h];
  }
  if (t < 2) plan->b2l1[t] = b2_1[jsh[t]];
}

// ---------------------------------------------------------------------------
// Main streaming kernel. Per wave: 16 batch rows.
//   g[16x8] (gathered x cols) @ Wblk[8x48] via 6x V_WMMA_F32_16X16X4_F32,
//   bias folded into the accumulator, ReLU, LDS transpose, then per-lane
//   layer-0 reduction + layer-1 module + sigmoid.
// ---------------------------------------------------------------------------
#define LDS_ROW 50   // padded row stride (floats) to avoid 48|64 bank conflicts

__global__ __launch_bounds__(256) void moe_main_kernel(const float* __restrict__ x,
                                                       const Plan* __restrict__ plan,
                                                       float* __restrict__ out) {
  __shared__ float hbuf[WAVES_PER_BLOCK][16 * LDS_ROW];

  const int lane = threadIdx.x & 31;
  const int wave = threadIdx.x >> 5;
  const int lo   = lane & 15;        // row within tile / N within tile
  const int hi   = lane >> 4;        // K/M split per WMMA layout
  const int rowbase = (blockIdx.x * WAVES_PER_BLOCK + wave) * 16;
  const int row = rowbase + lo;

  // ---- gather A fragments: a0 covers K=0..3 (g0..g3), a1 covers K=4..7 ----
  // A 16x4 layout: VGPR0 = K0 (lanes 0-15) / K2 (lanes 16-31); VGPR1 = K1/K3.
  const float* xr = x + (size_t)row * D_IN;
  const int ca = plan->c[2 * hi + 0];
  const int cb = plan->c[2 * hi + 1];
  const int cc = plan->c[4 + 2 * hi + 0];
  const int cd = plan->c[4 + 2 * hi + 1];
  v2f a0, a1;
  a0.x = xr[ca]; a0.y = xr[cb];
  a1.x = xr[cc]; a1.y = xr[cd];

  // ---- 3 N-tiles x 2 K-steps of f32 WMMA, bias in accumulator ----
  v8f acc[3];
#pragma unroll
  for (int nt = 0; nt < 3; ++nt) {
    const int n = nt * 16 + lo;
    const float bias = plan->b1blk[n];
    v8f cin = {bias, bias, bias, bias, bias, bias, bias, bias};
    // B 4x16 layout: VGPR0 = K0 (lanes 0-15) / K2 (lanes 16-31); VGPR1 = K1/K3.
    v2f b0, b1v;
    b0.x  = plan->Wblk[(0 + 2 * hi + 0) * 48 + n];
    b0.y  = plan->Wblk[(0 + 2 * hi + 1) * 48 + n];
    b1v.x = plan->Wblk[(4 + 2 * hi + 0) * 48 + n];
    b1v.y = plan->Wblk[(4 + 2 * hi + 1) * 48 + n];
    v8f tmp = __builtin_amdgcn_wmma_f32_16x16x4_f32(false, a0, false, b0, (short)0, cin, false, false);
    acc[nt]  = __builtin_amdgcn_wmma_f32_16x16x4_f32(false, a1, false, b1v, (short)0, tmp, false, false);
  }

  // ---- ReLU + transpose through LDS (component i -> row i + 8*hi) ----
  float* ws = hbuf[wave];
#pragma unroll
  for (int nt = 0; nt < 3; ++nt) {
    const int col = nt * 16 + lo;
#pragma unroll
    for (int i = 0; i < 8; ++i) {
      const float h = acc[nt][i];
      ws[(i + 8 * hi) * LDS_ROW + col] = h > 0.0f ? h : 0.0f;
    }
  }
  __syncthreads();

  // ---- tail: lane (lo,hi) -> row lo, layer-0 modules {2hi,2hi+1},
  //      layer-1 module hi, output column hi ----
  const float* rowp = ws + lo * LDS_ROW + hi * 24;
  float u0 = plan->b2l0[2 * hi + 0];
  float u1 = plan->b2l0[2 * hi + 1];
#pragma unroll
  for (int h = 0; h < HID; ++h) {
    u0 = fmaf(rowp[h],       plan->w2blk[24 * hi + h],       u0);
    u1 = fmaf(rowp[HID + h], plan->w2blk[24 * hi + HID + h], u1);
  }
  float y = plan->b2l1[hi];
#pragma unroll
  for (int h = 0; h < HID; ++h) {
    float hp = fmaf(plan->W1l1[hi * 24 + h], u0,
               fmaf(plan->W1l1[hi * 24 + HID + h], u1, plan->b1l1[hi * HID + h]));
    hp = hp > 0.0f ? hp : 0.0f;
    y = fmaf(hp, plan->w2l1[hi * HID + h], y);
  }
  out[(size_t)row * 2 + hi] = 1.0f / (1.0f + __expf(-y));
}

// ---------------------------------------------------------------------------
// Launch
// ---------------------------------------------------------------------------
extern "C" void kernel_launch(void* const* d_in, const int* in_sizes, int n_in,
                              void* d_out, int out_size, void* d_ws, size_t ws_size,
                              hipStream_t stream) {
  const float* x      = (const float*)d_in[0];
  const int*   taskid = (const int*)  d_in[1];
  const float* emb0   = (const float*)d_in[2];
  const float* emb1   = (const float*)d_in[3];
  const float* embo   = (const float*)d_in[4];
  const float* W1_0   = (const float*)d_in[5];
  const float* b1_0   = (const float*)d_in[6];
  const float* W2_0   = (const float*)d_in[7];
  const float* b2_0   = (const float*)d_in[8];
  const float* W1_1   = (const float*)d_in[9];
  const float* b1_1   = (const float*)d_in[10];
  const float* W2_1   = (const float*)d_in[11];
  const float* b2_1   = (const float*)d_in[12];

  Plan*  plan = (Plan*)d_ws;
  float* out  = (float*)d_out;

  setup_plan_kernel<<<1, 64, 0, stream>>>(emb0, emb1, embo, taskid,
                                          W1_0, b1_0, W2_0, b2_0,
                                          W1_1, b1_1, W2_1, b2_1, plan);
  moe_main_kernel<<<NBLOCKS, 256, 0, stream>>>(x, plan, out);
}